// message_39968965656903
// MI455X (gfx1250) — compile-verified
//
#include <hip/hip_runtime.h>
#include <hip/hip_bf16.h>
#include <math.h>

typedef __attribute__((ext_vector_type(2))) float v2f;
typedef __attribute__((ext_vector_type(8))) float v8f;

#define FDIM 128
#define TF   384          // 3*F
#define PI_OVER_RCUT 0.6283185307179586f   // pi/5
#define RCUT 5.0f

static __device__ __forceinline__ v8f wmma_f32_k4(v2f a, v2f b, v8f c) {
  // D = A(16x4 f32) * B(4x16 f32) + C(16x16 f32)
  return __builtin_amdgcn_wmma_f32_16x16x4_f32(
      false, a, false, b, (short)0, c, false, false);
}

// ---------------------------------------------------------------- kernel 1
__global__ void zero_kernel(float* __restrict__ out, size_t n, float* __restrict__ ws) {
  size_t i = (size_t)blockIdx.x * blockDim.x + threadIdx.x;
  if (i == 0) ws[0] = 0.0f;
  for (; i < n; i += (size_t)gridDim.x * blockDim.x) out[i] = 0.0f;
}

// ---------------------------------------------------------------- kernel 2
// Global Frobenius norm of r: ws[0] = sum(r*r) over all E*3 elements.
__global__ void r2_reduce_kernel(const float* __restrict__ r, float* __restrict__ ws, int total) {
  float acc = 0.0f;
  for (int i = blockIdx.x * blockDim.x + threadIdx.x; i < total;
       i += gridDim.x * blockDim.x) {
    float x = r[i];
    acc += x * x;
  }
  #pragma unroll
  for (int off = 16; off > 0; off >>= 1)
    acc += __shfl_down(acc, off, 32);
  if ((threadIdx.x & 31) == 0) atomicAdd(ws, acc);
}

// ---------------------------------------------------------------- kernel 3
// One wave32 handles 16 edges. Weights pre-swizzled into LDS in WMMA B-layout:
//   pair index p = (ntile*KC + c)*32 + lane ; floats [2p],[2p+1] hold
//   W[(4c + 2*(lane>>4) + {0,1})*384 + ntile*16 + (lane&15)]
__global__ __launch_bounds__(256) void painn_main_kernel(
    const float* __restrict__ s, const float* __restrict__ r,
    const float* __restrict__ v, const int* __restrict__ idx,
    const float* __restrict__ W_phi, const float* __restrict__ b_phi,
    const float* __restrict__ W_w, const float* __restrict__ b_w,
    float* __restrict__ out, const float* __restrict__ ws, int E, int N) {
  extern __shared__ float lds[];
  float* ldsPhi = lds;                  // 24*32*32*2 = 49152 floats (192 KB)
  float* ldsW   = lds + 24 * 32 * 64;   // 24*5*32*2  =  7680 floats (30 KB)

  const int tid = threadIdx.x;

  // ---- stage W_phi (K=128 -> 32 chunks of 4) swizzled into B layout
  for (int p = tid; p < 24 * 32 * 32; p += blockDim.x) {
    int nt = p >> 10, rem = p & 1023;
    int c = rem >> 5, l = rem & 31;
    int row = 4 * c + 2 * (l >> 4);
    int col = nt * 16 + (l & 15);
    ldsPhi[2 * p]     = W_phi[row * TF + col];
    ldsPhi[2 * p + 1] = W_phi[(row + 1) * TF + col];
  }
  // ---- stage W_w (K=20 -> 5 chunks of 4)
  for (int p = tid; p < 24 * 5 * 32; p += blockDim.x) {
    int nt = p / 160, rem = p % 160;
    int c = rem >> 5, l = rem & 31;
    int row = 4 * c + 2 * (l >> 4);
    int col = nt * 16 + (l & 15);
    ldsW[2 * p]     = W_w[row * TF + col];
    ldsW[2 * p + 1] = W_w[(row + 1) * TF + col];
  }
  __syncthreads();

  const int lane = tid & 31;
  const int waveId = tid >> 5;
  const int hi = lane >> 4;     // half-wave select (K offset)
  const int m = lane & 15;      // A-row / B-column within tile
  const float invG = rsqrtf(ws[0]);   // 1 / ||r||_F  (global scalar)

  float* out_v = out;                        // (N,3,F)
  float* out_s = out + (size_t)N * 3 * FDIM; // (N,1,F)

  const int nGroups = (E + 15) >> 4;
  const int wavesTotal = gridDim.x * (blockDim.x >> 5);
  for (int g = blockIdx.x * (blockDim.x >> 5) + waveId; g < nGroups; g += wavesTotal) {
    const int e0 = g * 16;
    const int em = e0 + m;
    const bool okm = (em < E);

    // ---- A tile of s: 16 edges x 128 K, fp32 WMMA A layout (64 VGPRs)
    v2f a_s[32];
    #pragma unroll
    for (int c = 0; c < 32; ++c) {
      if (okm) a_s[c] = *(const v2f*)(s + (size_t)em * FDIM + 4 * c + 2 * hi);
      else     { a_s[c].x = 0.0f; a_s[c].y = 0.0f; }
    }

    // ---- RBF A tile (16 edges x 20 K -> 5 chunks)
    float rx = 0.f, ry = 0.f, rz = 0.f;
    if (okm) { rx = r[em * 3 + 0]; ry = r[em * 3 + 1]; rz = r[em * 3 + 2]; }
    const float rn = sqrtf(rx * rx + ry * ry + rz * rz);
    v2f a_r[5];
    #pragma unroll
    for (int c = 0; c < 5; ++c) {
      #pragma unroll
      for (int jj = 0; jj < 2; ++jj) {
        int k = 4 * c + 2 * hi + jj;            // 0..19
        float t = sinf((float)(k + 1) * PI_OVER_RCUT * rn) / rn;
        float rb = (t <= RCUT) ? 0.5f * (cosf(t * PI_OVER_RCUT) + 1.0f) : 0.0f;
        if (!okm) rb = 0.0f;
        if (jj == 0) a_r[c].x = rb; else a_r[c].y = rb;
      }
    }

    // ---- per-output-edge data (C/D row M = j + 8*hi)
    int   node_j[8];
    float rsc[8][3];
    #pragma unroll
    for (int j = 0; j < 8; ++j) {
      int e = e0 + j + 8 * hi;
      if (e < E) {
        node_j[j] = idx[e];
        rsc[j][0] = r[e * 3 + 0] * invG;
        rsc[j][1] = r[e * 3 + 1] * invG;
        rsc[j][2] = r[e * 3 + 2] * invG;
      } else {
        node_j[j] = -1;
        rsc[j][0] = rsc[j][1] = rsc[j][2] = 0.0f;
      }
    }

    // ---- 8 column blocks; per block compute sv (nt), ss (nt+8), sr (nt+16)
    for (int ct = 0; ct < 8; ++ct) {
      float splitv[3][8];
      #pragma unroll
      for (int which = 0; which < 3; ++which) {
        const int nt = ct + which * 8;
        v8f acc = {};
        const float* bp = ldsPhi + nt * 2048 + lane * 2;
        #pragma unroll
        for (int c = 0; c < 32; ++c) {
          v2f b = *(const v2f*)(bp + c * 64);
          acc = wmma_f32_k4(a_s[c], b, acc);
        }
        v8f accw = {};
        const float* bw = ldsW + nt * 320 + lane * 2;
        #pragma unroll
        for (int c = 0; c < 5; ++c) {
          v2f b = *(const v2f*)(bw + c * 64);
          accw = wmma_f32_k4(a_r[c], b, accw);
        }
        const float bphi = b_phi[nt * 16 + m];
        const float bww  = b_w[nt * 16 + m];
        #pragma unroll
        for (int j = 0; j < 8; ++j)
          splitv[which][j] = (accw[j] + bww) * (acc[j] + bphi);
      }

      const int col = ct * 16 + m;
      #pragma unroll
      for (int j = 0; j < 8; ++j) {
        if (node_j[j] >= 0) {
          const int e = e0 + j + 8 * hi;
          atomicAdd(&out_s[(size_t)node_j[j] * FDIM + col], splitv[1][j]);
          #pragma unroll
          for (int d = 0; d < 3; ++d) {
            float vv  = v[((size_t)e * 3 + d) * FDIM + col];
            float msg = splitv[0][j] * vv + splitv[2][j] * rsc[j][d];
            atomicAdd(&out_v[((size_t)node_j[j] * 3 + d) * FDIM + col], msg);
          }
        }
      }
    }
  }
}

// ----------------------------------------------------------------
extern "C" void kernel_launch(void* const* d_in, const int* in_sizes, int n_in,
                              void* d_out, int out_size, void* d_ws, size_t ws_size,
                              hipStream_t stream) {
  const float* s     = (const float*)d_in[0];
  const float* r     = (const float*)d_in[1];
  const float* v     = (const float*)d_in[2];
  const int*   idx   = (const int*)  d_in[3];
  const float* W_phi = (const float*)d_in[4];
  const float* b_phi = (const float*)d_in[5];
  const float* W_w   = (const float*)d_in[6];
  const float* b_w   = (const float*)d_in[7];
  float* out = (float*)d_out;
  float* ws  = (float*)d_ws;

  const int E = in_sizes[3];
  const int N = out_size / (4 * FDIM);   // out = (N,3,F) + (N,1,F)

  zero_kernel<<<1024, 256, 0, stream>>>(out, (size_t)out_size, ws);
  r2_reduce_kernel<<<512, 256, 0, stream>>>(r, ws, E * 3);

  const size_t ldsBytes = (size_t)(24 * 32 * 64 + 24 * 5 * 64) * sizeof(float); // 222 KB
  painn_main_kernel<<<1024, 256, ldsBytes, stream>>>(
      s, r, v, idx, W_phi, b_phi, W_w, b_w, out, ws, E, N);
}